// layers_52690658787520
// MI455X (gfx1250) — compile-verified
//
#include <hip/hip_runtime.h>

// GAT layer on gfx1250 (MI455X), f32 WMMA path.
// Row-softmax cancellation: out[i,h*64+d] = (adj@(e*mx))[i,...] / (adj@e)[i,h],
// e[h,j] = exp(mx[j].a_dst[h] - max).  a_origin cancels; attention == one GEMM.
// B matrices stored column-major so WMMA B fragments are single b64 loads;
// 16x64 register blocking per wave -> 4 WMMA per A-fragment load.

typedef float v2f __attribute__((ext_vector_type(2)));
typedef float v4f __attribute__((ext_vector_type(4)));
typedef float v8f __attribute__((ext_vector_type(8)));

#define N_NODES 4096
#define IN_F    512
#define HEADS   8
#define DH      64
#define OUT_F   512   // HEADS*DH
#define YCOLS   528   // 512 numerator cols + 8 e-cols + 8 zero pad

static __device__ __forceinline__ v8f wmma4(v8f acc, v2f a, v2f b) {
  return __builtin_amdgcn_wmma_f32_16x16x4_f32(false, a, false, b, (short)0, acc,
                                               false, false);
}

// Shared inner GEMM: wave computes 16 rows x (NT*16) cols.
// ar = A + row(lane)*lda + half*2   (row-major A, K contiguous)
// bc = B + (cb+ln)*ldcol + half*2   (column-major B, K contiguous per column)
template <int NT>
static __device__ __forceinline__ void gemm_loop(const float* __restrict__ ar,
                                                 const float* __restrict__ bc,
                                                 size_t ldcol, int K, v8f* acc) {
  for (int kc = 0; kc < K; kc += 64) {
    __builtin_prefetch(ar + kc + 128, 0, 3);   // global_prefetch_b8 on the A stream
#pragma unroll
    for (int kk = 0; kk < 64; kk += 4) {
      const int k = kc + kk;
      const v2f a = *(const v2f*)(ar + k);     // A[row, k+half*2 .. +1]
#pragma unroll
      for (int t2 = 0; t2 < NT; ++t2) {
        const v2f b = *(const v2f*)(bc + (size_t)t2 * 16 * ldcol + k);
        acc[t2] = wmma4(acc[t2], a, b);
      }
    }
  }
}

// ---- transpose W [H,512,64] -> Wt [H,64,512] (column-major per head) -------
__global__ void k_wt(const float* __restrict__ W, float* __restrict__ Wt) {
  const int idx = blockIdx.x * 256 + threadIdx.x;   // 8*512*64 elements
  const int h = idx >> 15, rem = idx & 32767;
  const int k = rem >> 6, c = rem & 63;
  Wt[((size_t)h << 15) + (size_t)c * IN_F + k] = W[idx];
}

// ---- GEMM1: mx = x @ W  -- grid (N/64, H), block 128 (4 waves x 16 rows) ---
__global__ void k_proj(const float* __restrict__ x, const float* __restrict__ Wt,
                       float* __restrict__ mx) {
  const int lane = threadIdx.x & 31, wave = threadIdx.x >> 5;
  const int half = lane >> 4, ln = lane & 15;
  const int i0 = blockIdx.x * 64 + wave * 16;
  const int h  = blockIdx.y;
  const float* ar = x + (size_t)(i0 + ln) * IN_F + half * 2;
  const float* bc = Wt + ((size_t)h << 15) + (size_t)ln * IN_F + half * 2;
  v8f acc[4] = {};
  gemm_loop<4>(ar, bc, IN_F, IN_F, acc);
#pragma unroll
  for (int t2 = 0; t2 < 4; ++t2) {
    const int col = h * DH + t2 * 16 + ln;
#pragma unroll
    for (int r = 0; r < 8; ++r)
      mx[(size_t)(i0 + r + (half << 3)) * OUT_F + col] = acc[t2][r];
  }
}

// ---- t[h,j] = mx[j, h*64:] . a_dst[h] --------------------------------------
__global__ void k_tvec(const float* __restrict__ mx, const float* __restrict__ a_dst,
                       float* __restrict__ T) {
  const int idx = blockIdx.x * blockDim.x + threadIdx.x;  // N*H
  const int h = idx & (HEADS - 1);
  const int j = idx >> 3;
  const float* row = mx + (size_t)j * OUT_F + h * DH;
  const float* av  = a_dst + h * DH;
  float s = 0.f;
#pragma unroll
  for (int d = 0; d < DH; d += 4)
    s += row[d] * av[d] + row[d + 1] * av[d + 1]
       + row[d + 2] * av[d + 2] + row[d + 3] * av[d + 3];
  T[h * N_NODES + j] = s;
}

// ---- per-head max over j ---------------------------------------------------
__global__ void k_hmax(const float* __restrict__ T, float* __restrict__ maxv) {
  __shared__ float sm[256];
  const int h = blockIdx.x;
  float m = -3.4e38f;
  for (int j = threadIdx.x; j < N_NODES; j += 256)
    m = fmaxf(m, T[h * N_NODES + j]);
  sm[threadIdx.x] = m;
  __syncthreads();
  for (int s = 128; s > 0; s >>= 1) {
    if (threadIdx.x < s) sm[threadIdx.x] = fmaxf(sm[threadIdx.x], sm[threadIdx.x + s]);
    __syncthreads();
  }
  if (threadIdx.x == 0) maxv[h] = sm[0];
}

// ---- Yt[c, j] = e[h(c), j] * mx[j, c]  (column-major build + transpose) ----
// grid (N/16, 2), block 256: 16 rows x 256 cols per block; coalesced reads,
// each thread writes 16 consecutive k's of one column as 4x b128.
__global__ void k_buildT(const float* __restrict__ mx, const float* __restrict__ T,
                         const float* __restrict__ maxv, float* __restrict__ Yt) {
  __shared__ float esh[HEADS * 16];
  const int j0 = blockIdx.x * 16;
  const int c0 = blockIdx.y * 256;
  const int t  = threadIdx.x;
  if (t < HEADS * 16) {
    const int hh = t >> 4, jj = t & 15;
    esh[t] = __expf(T[hh * N_NODES + j0 + jj] - maxv[hh]);
  }
  __syncthreads();
  const int c = c0 + t;
  const int hh = c >> 6;
  float v[16];
#pragma unroll
  for (int jj = 0; jj < 16; ++jj)
    v[jj] = mx[(size_t)(j0 + jj) * OUT_F + c] * esh[hh * 16 + jj];
  float* dst = Yt + (size_t)c * N_NODES + j0;
#pragma unroll
  for (int q = 0; q < 4; ++q) {
    v4f p = {v[4 * q], v[4 * q + 1], v[4 * q + 2], v[4 * q + 3]};
    *(v4f*)(dst + 4 * q) = p;
  }
}

// ---- e-columns: Yt[512+cc, j] = (cc<8) ? e[cc,j] : 0 -----------------------
__global__ void k_ecols(const float* __restrict__ T, const float* __restrict__ maxv,
                        float* __restrict__ Yt) {
  const int j = blockIdx.x * 256 + threadIdx.x;
  const int cc = blockIdx.y;                 // 0..15
  const float val = (cc < HEADS) ? __expf(T[cc * N_NODES + j] - maxv[cc]) : 0.f;
  Yt[(size_t)(OUT_F + cc) * N_NODES + j] = val;
}

// ---- GEMM2: [Rnum|Rden] = adj @ Yt  -- grid (N/64, 9), block 128 -----------
__global__ void k_attn(const float* __restrict__ adj, const float* __restrict__ Yt,
                       float* __restrict__ rnum, float* __restrict__ rden) {
  const int lane = threadIdx.x & 31, wave = threadIdx.x >> 5;
  const int half = lane >> 4, ln = lane & 15;
  const int i0 = blockIdx.x * 64 + wave * 16;
  const int g  = blockIdx.y;
  const float* ar = adj + (size_t)(i0 + ln) * N_NODES + half * 2;
  if (g < 8) {                                        // numerator: 64 cols
    const int cb = g * 64;
    const float* bc = Yt + (size_t)(cb + ln) * N_NODES + half * 2;
    v8f acc[4] = {};
    gemm_loop<4>(ar, bc, N_NODES, N_NODES, acc);
#pragma unroll
    for (int t2 = 0; t2 < 4; ++t2)
#pragma unroll
      for (int r = 0; r < 8; ++r)
        rnum[(size_t)(i0 + r + (half << 3)) * OUT_F + cb + t2 * 16 + ln] = acc[t2][r];
  } else {                                            // denominator: 16 cols
    const float* bc = Yt + (size_t)(OUT_F + ln) * N_NODES + half * 2;
    v8f acc[1] = {};
    gemm_loop<1>(ar, bc, N_NODES, N_NODES, acc);
#pragma unroll
    for (int r = 0; r < 8; ++r)
      rden[(size_t)(i0 + r + (half << 3)) * 16 + ln] = acc[0][r];
  }
}

// ---- out = Rnum / Rden -----------------------------------------------------
__global__ void k_final(const float* __restrict__ rnum, const float* __restrict__ rden,
                        float* __restrict__ out) {
  const int idx = blockIdx.x * blockDim.x + threadIdx.x;  // N*512
  const int i = idx >> 9;
  const int h = (idx & 511) >> 6;
  out[idx] = rnum[idx] / rden[i * 16 + h];
}

extern "C" void kernel_launch(void* const* d_in, const int* in_sizes, int n_in,
                              void* d_out, int out_size, void* d_ws, size_t ws_size,
                              hipStream_t stream) {
  const float* x     = (const float*)d_in[0];
  const float* adj   = (const float*)d_in[1];
  const float* W     = (const float*)d_in[2];
  // d_in[3] = a_origin: cancels in the row softmax, not needed.
  const float* a_dst = (const float*)d_in[4];
  float* out = (float*)d_out;

  float* ws   = (float*)d_ws;
  float* MX   = ws;                                   // N*512  (reused as RNUM)
  float* T    = MX + (size_t)N_NODES * OUT_F;         // H*N
  float* MAXV = T + (size_t)HEADS * N_NODES;          // 16
  float* WT   = MAXV + 16;                            // H*64*512
  float* YT   = WT + (size_t)HEADS * DH * IN_F;       // YCOLS*N
  float* RD   = YT + (size_t)YCOLS * N_NODES;         // N*16
  float* RN   = MX;                                   // mx dead after k_buildT

  k_wt    <<<(HEADS * IN_F * DH) / 256, 256, 0, stream>>>(W, WT);
  k_proj  <<<dim3(N_NODES / 64, HEADS), 128, 0, stream>>>(x, WT, MX);
  k_tvec  <<<(N_NODES * HEADS) / 256, 256, 0, stream>>>(MX, a_dst, T);
  k_hmax  <<<HEADS, 256, 0, stream>>>(T, MAXV);
  k_buildT<<<dim3(N_NODES / 16, OUT_F / 256), 256, 0, stream>>>(MX, T, MAXV, YT);
  k_ecols <<<dim3(N_NODES / 256, 16), 256, 0, stream>>>(T, MAXV, YT);
  k_attn  <<<dim3(N_NODES / 64, 9), 128, 0, stream>>>(adj, YT, RN, RD);
  k_final <<<(N_NODES * OUT_F) / 256, 256, 0, stream>>>(RN, RD, out);
}